// EdgeGuidedFusionLoss_25529285607895
// MI455X (gfx1250) — compile-verified
//
#include <hip/hip_runtime.h>
#include <math.h>

typedef __attribute__((ext_vector_type(2))) float v2f;
typedef __attribute__((ext_vector_type(8))) float v8f;

#define BATCH 2
#define CH 3
#define IMH 256
#define IMW 256
#define HW (IMH*IMW)
#define PH 128
#define PW 128
#define PHW (PH*PW)
#define NSUB 4096
#define SINK_EPS 0.0025f
#define SINK_ITERS 11
#define SSIM_O 246
#define SSIM_N (SSIM_O*SSIM_O)

// ---- accumulator slots (floats at start of ws) ----
#define S_SSIM_M 0   // 0,1  per-b sums, ssim(y_gray, mri)
#define S_SSIM_P 2   // 2,3  per-b sums, ssim(y, pet)
#define S_SSIM_K 4   // 4,5  per-b sums, masked ssim
#define S_CH1 6      // charbonnier sobel full-res
#define S_CH2 7      // charbonnier sobel pooled
#define S_AUX 8
#define S_INT 9
#define S_SMOOTH 10
#define S_LOUT 11    // 11,12
#define S_CNT 13     // 13,14 mask>0.1 counts
#define S_MAX_YG 15  // 15,16 sobel max (int bits)
#define S_MAX_MRI 17 // 17,18
#define S_MAX_PY 19  // 19,20
#define S_MAX_PM 21  // 21,22
#define S_OT 24      // 24..29 OT values per problem
#define ACC_N 64

// ---- buffer offsets (floats) ----
#define O_YGRAY  64
#define O_POOLY  (O_YGRAY  + BATCH*HW)
#define O_POOLM  (O_POOLY  + BATCH*PHW)
#define O_SOB_YG (O_POOLM  + BATCH*PHW)
#define O_SOB_MRI (O_SOB_YG + BATCH*HW)
#define O_SOB_PY (O_SOB_MRI + BATCH*HW)
#define O_SOB_PM (O_SOB_PY + BATCH*PHW)
#define O_YPTS   (O_SOB_PM + BATCH*PHW)     // float4[B*NSUB]
#define O_PPTS   (O_YPTS   + BATCH*NSUB*4)  // float4[B*NSUB]
#define O_SQX    (O_PPTS   + BATCH*NSUB*4)
#define O_SQY    (O_SQX    + BATCH*NSUB)
#define O_FG     (O_SQY    + BATCH*NSUB)    // [6][2][NSUB]
#define FG_N (6*2*NSUB)

static inline int cdiv(int a, int b) { return (a + b - 1) / b; }

// ---------------- device helpers ----------------
__device__ __forceinline__ float blockSum(float v) {
  __shared__ float red[256];
  int t = threadIdx.x;
  red[t] = v; __syncthreads();
  for (int s = 128; s > 0; s >>= 1) { if (t < s) red[t] += red[t + s]; __syncthreads(); }
  float r = red[0]; __syncthreads();
  return r;
}
__device__ __forceinline__ float blockMax(float v) {
  __shared__ float redm[256];
  int t = threadIdx.x;
  redm[t] = v; __syncthreads();
  for (int s = 128; s > 0; s >>= 1) { if (t < s) redm[t] = fmaxf(redm[t], redm[t + s]); __syncthreads(); }
  float r = redm[0]; __syncthreads();
  return r;
}
__device__ __forceinline__ void gauss1d(float* k, int n, float sigma) {
  float s = 0.f;
  for (int i = 0; i < n; ++i) {
    float x = (float)i - (float)(n - 1) * 0.5f;
    k[i] = __expf(-(x * x) / (2.f * sigma * sigma));
    s += k[i];
  }
  float inv = 1.f / s;
  for (int i = 0; i < n; ++i) k[i] *= inv;
}
__device__ __forceinline__ int mirror(int p, int n) {
  if (p < 0) p = -p - 1;
  if (p >= n) p = 2 * n - 1 - p;
  return p;
}

// ---------------- small prep kernels ----------------
__global__ void k_init(float* ws) {
  int t = blockIdx.x * blockDim.x + threadIdx.x;
  if (t < ACC_N) ws[t] = 0.f;
  int f = t - ACC_N;
  if (f >= 0 && f < FG_N) ws[O_FG + f] = 0.f;
}

__global__ void k_gray(const float* __restrict__ y, float* ws) {
  int t = blockIdx.x * blockDim.x + threadIdx.x;
  if (t >= BATCH * HW) return;
  int b = t / HW, i = t - b * HW;
  ws[O_YGRAY + t] = (y[(size_t)(b * 3 + 0) * HW + i] + y[(size_t)(b * 3 + 1) * HW + i] +
                     y[(size_t)(b * 3 + 2) * HW + i]) * (1.f / 3.f);
}

__global__ void k_pool(const float* __restrict__ mri, float* ws) {
  int t = blockIdx.x * blockDim.x + threadIdx.x;
  if (t >= BATCH * PHW) return;
  int b = t / PHW, r = t - b * PHW, ph = r / PW, pw = r - ph * PW;
  int h = 2 * ph, w = 2 * pw;
  const float* yg = ws + O_YGRAY + (size_t)b * HW;
  ws[O_POOLY + t] = 0.25f * (yg[h * IMW + w] + yg[h * IMW + w + 1] +
                             yg[(h + 1) * IMW + w] + yg[(h + 1) * IMW + w + 1]);
  const float* mb = mri + (size_t)b * HW;
  ws[O_POOLM + t] = 0.25f * (mb[h * IMW + w] + mb[h * IMW + w + 1] +
                             mb[(h + 1) * IMW + w] + mb[(h + 1) * IMW + w + 1]);
}

// ---------------- sobel magnitude + per-image max ----------------
__global__ void k_sobel(const float* __restrict__ img, float* __restrict__ mag,
                        int maxSlot, int Hi, int Wi, float* ws) {
  int b = blockIdx.y;
  int t = blockIdx.x * blockDim.x + threadIdx.x;
  int n = Hi * Wi;
  float mval = 0.f;
  if (t < n) {
    int h = t / Wi, w = t - h * Wi;
    const float* ib = img + (size_t)b * n;
    const float kx[3][3] = {{1, 0, -1}, {2, 0, -2}, {1, 0, -1}};
    const float ky[3][3] = {{1, 2, 1}, {0, 0, 0}, {-1, -2, -1}};
    float gx = 0.f, gy = 0.f;
    for (int u = 0; u < 3; ++u)
      for (int v = 0; v < 3; ++v) {
        int hh = h + u - 1, wj = w + v - 1;
        float x = (hh >= 0 && hh < Hi && wj >= 0 && wj < Wi) ? ib[hh * Wi + wj] : 0.f;
        gx += kx[u][v] * x; gy += ky[u][v] * x;
      }
    mval = sqrtf(gx * gx + gy * gy + 1e-6f);
    mag[(size_t)b * n + t] = mval;
  }
  float mred = blockMax(mval);
  if (threadIdx.x == 0) atomicMax((int*)(ws + maxSlot + b), __float_as_int(mred));
}

__global__ void k_charb(const float* __restrict__ magA, int slotA,
                        const float* __restrict__ magB, int slotB,
                        int n, int accSlot, float* ws) {
  int b = blockIdx.y;
  int t = blockIdx.x * blockDim.x + threadIdx.x;
  float v = 0.f;
  if (t < n) {
    float ma = fmaxf(__int_as_float(((const int*)ws)[slotA + b]), 1e-6f);
    float mb = fmaxf(__int_as_float(((const int*)ws)[slotB + b]), 1e-6f);
    float d = magA[(size_t)b * n + t] / ma - magB[(size_t)b * n + t] / mb;
    v = sqrtf(d * d + 1e-6f);
  }
  float s = blockSum(v);
  if (threadIdx.x == 0) atomicAdd(ws + accSlot, s);
}

__global__ void k_aux(const float* __restrict__ ehat, const float* __restrict__ magM, float* ws) {
  int b = blockIdx.y;
  int t = blockIdx.x * blockDim.x + threadIdx.x;
  float v = 0.f;
  if (t < HW) {
    float mm = fmaxf(__int_as_float(((const int*)ws)[S_MAX_MRI + b]), 1e-6f);
    v = fabsf(ehat[(size_t)b * HW + t] - magM[(size_t)b * HW + t] / mm);
  }
  float s = blockSum(v);
  if (threadIdx.x == 0) atomicAdd(ws + S_AUX, s);
}

// ---------------- pixelwise terms: intensity, L_out, mask count ----------------
__global__ void k_pix(const float* __restrict__ y, const float* __restrict__ pet,
                      const float* __restrict__ mask, float* ws) {
  int b = blockIdx.y;
  int t = blockIdx.x * blockDim.x + threadIdx.x;
  float ai = 0.f, ao = 0.f, ac = 0.f;
  if (t < CH * HW) {
    int c = t / HW, i = t - c * HW;
    float m = mask[(size_t)b * HW + i];
    float yv = y[((size_t)(b * CH + c)) * HW + i];
    float pv = pet[((size_t)(b * CH + c)) * HW + i];
    float d = (yv - pv) * m;
    ai = d * d;
    float o = yv * (1.f - m);
    ao = o * o;
    if (c == 0) ac = (m > 0.1f) ? 1.f : 0.f;
  }
  float s1 = blockSum(ai);
  if (threadIdx.x == 0) atomicAdd(ws + S_INT, s1);
  float s2 = blockSum(ao);
  if (threadIdx.x == 0) atomicAdd(ws + S_LOUT + b, s2);
  float s3 = blockSum(ac);
  if (threadIdx.x == 0) atomicAdd(ws + S_CNT + b, s3);
}

// ---------------- gaussian smooth (sigma=1, radius 4, symmetric pad) ----------------
__global__ void k_smooth(const float* __restrict__ y, float* ws) {
  int b = blockIdx.y;
  int t = blockIdx.x * blockDim.x + threadIdx.x;
  float acc = 0.f;
  if (t < CH * HW) {
    int c = t / HW, r = t - c * HW, h = r / IMW, w = r - h * IMW;
    float k[9];
    gauss1d(k, 9, 1.0f);
    const float* yb = y + ((size_t)(b * CH + c)) * HW;
    float s = 0.f;
    for (int u = 0; u < 9; ++u) {
      int hh = mirror(h + u - 4, IMH);
      float rs = 0.f;
      for (int v = 0; v < 9; ++v) {
        int wj = mirror(w + v - 4, IMW);
        rs += k[v] * yb[hh * IMW + wj];
      }
      s += k[u] * rs;
    }
    float d = s - yb[h * IMW + w];
    acc = d * d;
  }
  float sr = blockSum(acc);
  if (threadIdx.x == 0) atomicAdd(ws + S_SMOOTH, sr);
}

// ---------------- SSIM (11x11 gaussian window, VALID) ----------------
__global__ void k_ssim(const float* __restrict__ A, const float* __restrict__ Bi,
                       const float* __restrict__ mask, int Cch, int bBcast,
                       int accSlot, float* ws) {
  int b = blockIdx.y;
  int t = blockIdx.x * blockDim.x + threadIdx.x;
  int tot = Cch * SSIM_N;
  float val = 0.f;
  if (t < tot) {
    int c = t / SSIM_N, r2 = t - c * SSIM_N, oh = r2 / SSIM_O, ow = r2 - oh * SSIM_O;
    float k[11];
    gauss1d(k, 11, 1.5f);
    const float* Ab = A + ((size_t)(b * Cch + c)) * HW;
    const float* Bb = Bi + ((size_t)(b * (bBcast ? 1 : Cch) + (bBcast ? 0 : c))) * HW;
    const float* Mb = mask ? (mask + (size_t)b * HW) : (const float*)0;
    float sx = 0.f, sy = 0.f, sxx = 0.f, syy = 0.f, sxy = 0.f;
    for (int u = 0; u < 11; ++u) {
      int h = oh + u;
      for (int v = 0; v < 11; ++v) {
        int w0 = ow + v;
        float wgt = k[u] * k[v];
        float mm = Mb ? Mb[h * IMW + w0] : 1.f;
        float xv = Ab[h * IMW + w0] * mm;
        float yv = Bb[h * IMW + w0] * mm;
        sx += wgt * xv; sy += wgt * yv;
        sxx += wgt * xv * xv; syy += wgt * yv * yv; sxy += wgt * xv * yv;
      }
    }
    float s11 = sxx - sx * sx, s22 = syy - sy * sy, s12 = sxy - sx * sy;
    const float C1 = 1e-4f, C2 = 9e-4f;
    val = ((2.f * sx * sy + C1) * (2.f * s12 + C2)) /
          ((sx * sx + sy * sy + C1) * (s11 + s22 + C2));
  }
  float s = blockSum(val);
  if (threadIdx.x == 0) atomicAdd(ws + accSlot + b, s);
}

// ---------------- gather masked point clouds ----------------
__global__ void k_gather(const float* __restrict__ y, const float* __restrict__ pet,
                         const float* __restrict__ mask, const int* __restrict__ idx,
                         float* ws) {
  int t = blockIdx.x * blockDim.x + threadIdx.x;
  if (t >= BATCH * NSUB) return;
  int b = t / NSUB;
  int i = idx[t];
  float m = mask[(size_t)b * HW + i];
  float a0 = y[(size_t)(b * 3 + 0) * HW + i] * m;
  float a1 = y[(size_t)(b * 3 + 1) * HW + i] * m;
  float a2 = y[(size_t)(b * 3 + 2) * HW + i] * m;
  float p0 = pet[(size_t)(b * 3 + 0) * HW + i] * m;
  float p1 = pet[(size_t)(b * 3 + 1) * HW + i] * m;
  float p2 = pet[(size_t)(b * 3 + 2) * HW + i] * m;
  ((float4*)(ws + O_YPTS))[t] = make_float4(a0, a1, a2, 0.f);
  ((float4*)(ws + O_PPTS))[t] = make_float4(p0, p1, p2, 0.f);
  ws[O_SQX + t] = a0 * a0 + a1 * a1 + a2 * a2;
  ws[O_SQY + t] = p0 * p0 + p1 * p1 + p2 * p2;
}

// ---------------- Sinkhorn half-iteration via V_WMMA_F32_16X16X4_F32 ----------------
// out[j] = 0.5*|q_j|^2 - eps*(log(1/N) + LSE_i( dot(p_i, q_j/eps) + r_i/eps )),
// r_i = fin[i] - 0.5*|p_i|^2. The r/eps term rides in the K=3 slot of the A operand
// (patched into LDS after an async bulk copy); B columns carry (q/eps, 1), so the
// WMMA output IS the LSE argument — no per-tile rescale needed.
__global__ void k_sink_half(float* ws, int phase) {
  const float4* ypts = (const float4*)(ws + O_YPTS);
  const float4* ppts = (const float4*)(ws + O_PPTS);
  const float* sqxA = ws + O_SQX;
  const float* sqyA = ws + O_SQY;
  int p = blockIdx.y;
  int b = p / 3, kind = p - 3 * b;
  const float4* X = (kind == 2) ? (ppts + b * NSUB) : (ypts + b * NSUB);
  const float4* Y = (kind == 1) ? (ypts + b * NSUB) : (ppts + b * NSUB);
  const float* sqX = (kind == 2) ? (sqyA + b * NSUB) : (sqxA + b * NSUB);
  const float* sqY = (kind == 1) ? (sqxA + b * NSUB) : (sqyA + b * NSUB);
  float* f = ws + O_FG + p * 2 * NSUB;
  float* g = f + NSUB;

  const float4* P; const float* sqP; const float* fin;
  const float4* Q; const float* sqQ; float* out;
  if (phase == 0) { P = X; sqP = sqX; fin = f; Q = Y; sqQ = sqY; out = g; }
  else           { P = Y; sqP = sqY; fin = g; Q = X; sqQ = sqX; out = f; }

  extern __shared__ float4 Plds[];  // 4096 float4 = 64KB dynamic LDS
  const float inv_eps = 1.0f / SINK_EPS;

  // Stage raw P points into LDS with the CDNA5 async copy path (ASYNCcnt),
  // then patch the .w slot with the per-row LSE offset (f_i - 0.5|p_i|^2)/eps.
  unsigned ldsBase = __builtin_amdgcn_groupstaticsize();  // dynamic-LDS base offset
  for (int i = threadIdx.x; i < NSUB; i += blockDim.x) {
    unsigned laddr = ldsBase + (unsigned)i * 16u;
    unsigned long long ga = (unsigned long long)(const void*)&P[i];
    asm volatile("global_load_async_to_lds_b128 %0, %1, off"
                 :: "v"(laddr), "v"(ga) : "memory");
  }
  asm volatile("s_wait_asynccnt 0x0" ::: "memory");
  for (int i = threadIdx.x; i < NSUB; i += blockDim.x) {
    ((float*)(&Plds[i]))[3] = (fin[i] - 0.5f * sqP[i]) * inv_eps;
  }
  __syncthreads();

  int lane = threadIdx.x & 31;
  int wv = threadIdx.x >> 5;
  int jbase = (blockIdx.x * (blockDim.x >> 5) + wv) * 16;
  int lmod = lane & 15, lhalf = lane >> 4;
  int j = jbase + lmod;

  // B operand (4x16): VGPR0 lanes0-15 K=0 row, lanes16-31 K=2 row; VGPR1 K=1/K=3.
  // Columns hold (qx/eps, qy/eps, qz/eps, 1).
  float4 qv = Q[j];
  v2f bq;
  bq.x = (lhalf ? qv.z : qv.x) * inv_eps;
  bq.y = lhalf ? 1.0f : (qv.y * inv_eps);

  const float2* P2 = (const float2*)Plds;
  // 8 independent online-LSE chains (one per D VGPR) for exp-latency hiding.
  float mx[8], sm[8];
#pragma unroll
  for (int r = 0; r < 8; ++r) { mx[r] = -1e30f; sm[r] = 0.f; }

#pragma unroll 2
  for (int it = 0; it < NSUB / 16; ++it) {
    int ibase = it * 16;
    // A operand (16x4): lanes0-15 coords (x,y) of row i, lanes16-31 coords (z, r/eps).
    float2 av = P2[((ibase + lmod) << 1) + lhalf];
    v2f a; a.x = av.x; a.y = av.y;
    v8f cz = {0.f, 0.f, 0.f, 0.f, 0.f, 0.f, 0.f, 0.f};
    v8f d = __builtin_amdgcn_wmma_f32_16x16x4_f32(false, a, false, bq,
                                                  (short)0, cz, false, false);
#pragma unroll
    for (int r = 0; r < 8; ++r) {
      float t = d[r];
      float nm = fmaxf(mx[r], t);
      sm[r] = sm[r] * __expf(mx[r] - nm) + __expf(t - nm);
      mx[r] = nm;
    }
  }
  // merge the 8 chains
  float M = mx[0], S = sm[0];
#pragma unroll
  for (int r = 1; r < 8; ++r) {
    float nm = fmaxf(M, mx[r]);
    S = S * __expf(M - nm) + sm[r] * __expf(mx[r] - nm);
    M = nm;
  }
  // merge lane halves: lanes0-15 hold rows 0-7 of column j, lanes16-31 rows 8-15
  float mo = __shfl_xor(M, 16);
  float so = __shfl_xor(S, 16);
  float Mm = fmaxf(M, mo);
  float Ss = S * __expf(M - Mm) + so * __expf(mo - Mm);
  if (lane < 16) {
    const float logw = -8.3177661667193433f;  // -ln(4096)
    out[j] = 0.5f * sqQ[j] - SINK_EPS * (logw + Mm + __logf(Ss));
  }
}

__global__ void k_otval(float* ws) {
  int p = blockIdx.x;
  const float* fg = ws + O_FG + p * 2 * NSUB;
  float s = 0.f;
  for (int i = threadIdx.x; i < 2 * NSUB; i += blockDim.x) s += fg[i];
  float red = blockSum(s);
  if (threadIdx.x == 0) ws[S_OT + p] = red / (float)NSUB;  // mean(f)+mean(g)
}

// ---------------- final assembly ----------------
__global__ void k_final(const float* ws, float* out) {
  if (threadIdx.x != 0 || blockIdx.x != 0) return;
  float inc[2], sdiv[2];
  for (int b = 0; b < 2; ++b) {
    inc[b] = (ws[S_CNT + b] >= 32.f) ? 1.f : 0.f;
    sdiv[b] = ws[S_OT + 3 * b] - 0.5f * ws[S_OT + 3 * b + 1] - 0.5f * ws[S_OT + 3 * b + 2];
  }
  float Lt = 0.f, Lo = 0.f, Ls = 0.f;
  const float ssimN = (float)SSIM_N;
  for (int b = 0; b < 2; ++b) {
    Lt += inc[b] * sdiv[b];
    Lo += inc[b] * (ws[S_LOUT + b] / (3.f * (float)HW));
    Ls += inc[b] * (1.f - ws[S_SSIM_K + b] / (3.f * ssimN));
  }
  Lt *= 0.5f; Lo *= 0.5f; Ls *= 0.5f;
  float loss_transport = 0.8f * Lt + 0.3f * Lo + 0.5f * Ls;
  float L_int = 2.5f * ws[S_INT] / (2.f * 3.f * (float)HW);
  float L_sm = ws[S_SMOOTH] / (2.f * 3.f * (float)HW);
  float total = loss_transport + L_int + 0.2f * L_sm;
  float ssim_m = 1.f - (ws[S_SSIM_M] + ws[S_SSIM_M + 1]) / (2.f * ssimN);
  float ssim_p = 1.f - (ws[S_SSIM_P] + ws[S_SSIM_P + 1]) / (2.f * 3.f * ssimN);
  float g_cons = ws[S_CH1] / (2.f * (float)HW) + ws[S_CH2] / (2.f * (float)PHW);
  float aux = ws[S_AUX] / (2.f * (float)HW);
  out[0] = total; out[1] = ssim_m; out[2] = ssim_p; out[3] = g_cons; out[4] = aux;
}

// ---------------- host launcher ----------------
extern "C" void kernel_launch(void* const* d_in, const int* in_sizes, int n_in,
                              void* d_out, int out_size, void* d_ws, size_t ws_size,
                              hipStream_t stream) {
  (void)in_sizes; (void)n_in; (void)out_size; (void)ws_size;
  const float* y    = (const float*)d_in[0];
  const float* mri  = (const float*)d_in[1];
  const float* pet  = (const float*)d_in[2];
  const float* mask = (const float*)d_in[3];
  const float* ehat = (const float*)d_in[4];
  const int*   idx  = (const int*)d_in[5];
  float* ws = (float*)d_ws;
  float* outp = (float*)d_out;
  const dim3 blk(256);

  k_init<<<cdiv(ACC_N + FG_N, 256), blk, 0, stream>>>(ws);
  k_gray<<<cdiv(BATCH * HW, 256), blk, 0, stream>>>(y, ws);
  k_pool<<<cdiv(BATCH * PHW, 256), blk, 0, stream>>>(mri, ws);

  k_sobel<<<dim3(cdiv(HW, 256), BATCH), blk, 0, stream>>>(ws + O_YGRAY, ws + O_SOB_YG, S_MAX_YG, IMH, IMW, ws);
  k_sobel<<<dim3(cdiv(HW, 256), BATCH), blk, 0, stream>>>(mri, ws + O_SOB_MRI, S_MAX_MRI, IMH, IMW, ws);
  k_sobel<<<dim3(cdiv(PHW, 256), BATCH), blk, 0, stream>>>(ws + O_POOLY, ws + O_SOB_PY, S_MAX_PY, PH, PW, ws);
  k_sobel<<<dim3(cdiv(PHW, 256), BATCH), blk, 0, stream>>>(ws + O_POOLM, ws + O_SOB_PM, S_MAX_PM, PH, PW, ws);

  k_charb<<<dim3(cdiv(HW, 256), BATCH), blk, 0, stream>>>(ws + O_SOB_YG, S_MAX_YG, ws + O_SOB_MRI, S_MAX_MRI, HW, S_CH1, ws);
  k_charb<<<dim3(cdiv(PHW, 256), BATCH), blk, 0, stream>>>(ws + O_SOB_PY, S_MAX_PY, ws + O_SOB_PM, S_MAX_PM, PHW, S_CH2, ws);
  k_aux<<<dim3(cdiv(HW, 256), BATCH), blk, 0, stream>>>(ehat, ws + O_SOB_MRI, ws);

  k_pix<<<dim3(cdiv(CH * HW, 256), BATCH), blk, 0, stream>>>(y, pet, mask, ws);
  k_smooth<<<dim3(cdiv(CH * HW, 256), BATCH), blk, 0, stream>>>(y, ws);

  k_ssim<<<dim3(cdiv(SSIM_N, 256), BATCH), blk, 0, stream>>>(ws + O_YGRAY, mri, (const float*)0, 1, 0, S_SSIM_M, ws);
  k_ssim<<<dim3(cdiv(3 * SSIM_N, 256), BATCH), blk, 0, stream>>>(y, pet, (const float*)0, 3, 0, S_SSIM_P, ws);
  k_ssim<<<dim3(cdiv(3 * SSIM_N, 256), BATCH), blk, 0, stream>>>(y, mri, mask, 3, 1, S_SSIM_K, ws);

  k_gather<<<cdiv(BATCH * NSUB, 256), blk, 0, stream>>>(y, pet, mask, idx, ws);

  const size_t ldsBytes = (size_t)NSUB * sizeof(float4);  // 64KB
  for (int it = 0; it < SINK_ITERS; ++it) {
    k_sink_half<<<dim3(32, 6), blk, ldsBytes, stream>>>(ws, 0);  // g from f
    k_sink_half<<<dim3(32, 6), blk, ldsBytes, stream>>>(ws, 1);  // f from g
  }
  k_otval<<<6, blk, 0, stream>>>(ws);
  k_final<<<1, 64, 0, stream>>>(ws, outp);
}